// KB_2456721293922
// MI455X (gfx1250) — compile-verified
//
#include <hip/hip_runtime.h>

typedef float v2f __attribute__((ext_vector_type(2)));
typedef float v8f __attribute__((ext_vector_type(8)));

#define DDIM 128
#define TILE 16
#define WAVES_PER_BLOCK 4
#define LDS_STRIDE 34   // 32 lanes + 2 pad floats -> conflict-free b64 reads

// One wave handles 16 edges:
//  phase 1: per edge, all 32 lanes cooperatively load the 512B rows (coalesced
//           float4), each lane keeps a 4-element |h[r]+g[t]-h[c]| partial sum.
//  phase 2: partial matrix P[edge][srcLane] staged in LDS, reduced over the 32
//           source lanes with 8 chained V_WMMA_F32_16X16X4_F32 (B = ones).
//  phase 3: with B all-ones every D column is identical: lane0 holds d[0..7],
//           lane16 holds d[8..15] in the 8 accumulator VGPRs -> b128 stores.
__global__ __launch_bounds__(WAVES_PER_BLOCK * 32) void KB_transe_l1_wmma(
    const float* __restrict__ h,
    const float* __restrict__ g,
    const int* __restrict__ row,
    const int* __restrict__ col,
    const int* __restrict__ typ,
    float* __restrict__ out,
    int n_edges)
{
    __shared__ float lds[WAVES_PER_BLOCK * TILE * LDS_STRIDE];

    const int lane  = threadIdx.x & 31;
    const int wave  = threadIdx.x >> 5;
    const int tile  = blockIdx.x * WAVES_PER_BLOCK + wave;
    const int ebase = tile * TILE;
    if (ebase >= n_edges) return;

    const int nvalid = min(TILE, n_edges - ebase);
    const int eslot  = lane & 15;
    const int eclamp = ebase + min(eslot, nvalid - 1);

    // lanes 0..15 own the 16 edge descriptors (lanes 16..31 mirror them)
    int myRow = row[eclamp];
    int myCol = col[eclamp];
    int myTyp = typ[eclamp];

    float* P = &lds[wave * TILE * LDS_STRIDE];

    // ---- phase 1: coalesced gathers + per-lane partial L1 sums -------------
    #pragma unroll 4
    for (int e = 0; e < TILE; ++e) {
        const int r = __shfl(myRow, e, 32);
        const int c = __shfl(myCol, e, 32);
        const int t = __shfl(myTyp, e, 32);
        const float4 hr = reinterpret_cast<const float4*>(h + (size_t)r * DDIM)[lane];
        const float4 hc = reinterpret_cast<const float4*>(h + (size_t)c * DDIM)[lane];
        const float4 gv = reinterpret_cast<const float4*>(g + (size_t)t * DDIM)[lane];
        const float s = fabsf(hr.x + gv.x - hc.x)
                      + fabsf(hr.y + gv.y - hc.y)
                      + fabsf(hr.z + gv.z - hc.z)
                      + fabsf(hr.w + gv.w - hc.w);
        P[e * LDS_STRIDE + lane] = s;   // P[edge][srcLane], padded stride
    }
    __syncthreads();   // order LDS store -> transposed LDS read

    // ---- phase 2: WMMA reduction over the 32 source lanes ------------------
    // A-matrix (16x4 f32) layout: lane L holds (M=L%16, K=k0) in VGPR0 and
    // (M, k0+1) in VGPR1, with k0 = (L<16)?0:2.  A[m,k] := P[m][4*chunk + k].
    v8f acc = {};
    v2f ones; ones.x = 1.0f; ones.y = 1.0f;
    const int k0 = (lane >> 4) << 1;
    #pragma unroll
    for (int ch = 0; ch < 8; ++ch) {
        const float* src = &P[eslot * LDS_STRIDE + 4 * ch + k0];
        v2f a = *reinterpret_cast<const v2f*>(src);   // ds_load_b64, conflict-free
        acc = __builtin_amdgcn_wmma_f32_16x16x4_f32(
                  false, a, false, ones, (short)0, acc, false, false);
    }

    // ---- phase 3: extract column (all columns equal) and store -------------
    if (eslot == 0) {                 // lanes 0 and 16
        const int mbase = (lane >> 4) * 8;
        if (nvalid == TILE) {
            float4 lo = make_float4(acc[0], acc[1], acc[2], acc[3]);
            float4 hi = make_float4(acc[4], acc[5], acc[6], acc[7]);
            reinterpret_cast<float4*>(out + ebase + mbase)[0] = lo;
            reinterpret_cast<float4*>(out + ebase + mbase)[1] = hi;
        } else {
            for (int i = 0; i < 8; ++i) {
                const int m = mbase + i;
                if (m < nvalid) out[ebase + m] = acc[i];
            }
        }
    }
}

extern "C" void kernel_launch(void* const* d_in, const int* in_sizes, int n_in,
                              void* d_out, int out_size, void* d_ws, size_t ws_size,
                              hipStream_t stream) {
    const float* h    = (const float*)d_in[0];
    const float* g    = (const float*)d_in[1];
    const int*   eidx = (const int*)d_in[2];   // [2, E] flat: rows then cols
    const int*   etyp = (const int*)d_in[3];
    float*       out  = (float*)d_out;

    const int E     = in_sizes[3];                       // 600000
    const int tiles = (E + TILE - 1) / TILE;             // 37500
    const int grid  = (tiles + WAVES_PER_BLOCK - 1) / WAVES_PER_BLOCK;  // 9375

    KB_transe_l1_wmma<<<grid, WAVES_PER_BLOCK * 32, 0, stream>>>(
        h, g, eidx, eidx + E, etyp, out, E);
}